// Channel2D2_38817914421494
// MI455X (gfx1250) — compile-verified
//
#include <hip/hip_runtime.h>

typedef __attribute__((ext_vector_type(2))) float v2f;
typedef __attribute__((ext_vector_type(8))) float v8f;

// Vertex-membership bitmask per triangle (bit k set => vertex k is in triangle).
#define TM(a,b,c) ((1u<<(a))|(1u<<(b))|(1u<<(c)))
// Tile 0: triangles 0..15 in slots 0..15.
__constant__ unsigned kMaskA[16] = {
  TM(0,3,4),  TM(0,1,4),  TM(1,4,5),   TM(1,2,5),   TM(2,5,6),
  TM(3,4,7),  TM(4,7,8),  TM(4,5,8),   TM(5,8,9),   TM(5,6,9),
  TM(7,10,11),TM(7,8,11), TM(8,11,12), TM(8,9,12),  TM(9,12,13),
  TM(10,11,14)
};
// Tile 1: triangles 16..22 in slots 0..6, 23..29 in slots 8..14.
// Slots 7 and 15 are zero rows -> acc1 VGPR 7 is entirely dead (never stored),
// so every store in the loop is unconditional (no EXEC divergence).
__constant__ unsigned kMaskB[16] = {
  TM(11,14,15),TM(11,12,15),TM(12,15,16),TM(12,13,16),TM(14,17,18),
  TM(14,15,18),TM(15,18,19), 0u,
  TM(15,16,19),TM(16,19,20),TM(17,18,21),TM(18,21,22),TM(18,19,22),
  TM(19,22,23),TM(19,20,23), 0u
};
#undef TM

// out[slice, t, l] = sum_k W[t,k] * x[slice, k, l],  W[t,k] = (1/3) * (k in TRI[t])
// Two 16x16x4 f32 WMMA accumulator chains (triangle tiles) over six K-steps of 4,
// N = 16 columns of l per wave-tile. A fragments are compile-time constants.
__global__ __launch_bounds__(256) void tri_mean_wmma(const float* __restrict__ x,
                                                     float* __restrict__ out) {
  constexpr int  L = 32768;   // columns
  constexpr int  V = 24;      // vertices (K)
  constexpr int  T = 30;      // triangles (M)
  constexpr int  TILES_PER_WAVE = 8;
  constexpr float THIRD = 1.0f / 3.0f;

  const int lane  = threadIdx.x & 31;
  const int gw    = (int)((blockIdx.x * blockDim.x + threadIdx.x) >> 5); // global wave id
  const int slice = gw >> 8;                 // 64 (b*c) slices, 256 waves each
  const int widx  = gw & 255;
  const int col   = lane & 15;               // N index
  const int hf    = lane >> 4;               // half-wave selector

  // --- Constant A fragments (built once per wave, reused for its 8 tiles = 128 columns) ---
  // 16x4 f32 A layout: lanes 0-15 = M rows with K{4kk,4kk+1}; lanes 16-31 same rows, K{+2,+3}.
  const unsigned m0 = kMaskA[col];
  const unsigned m1 = kMaskB[col];
  v2f a0[6], a1[6];
#pragma unroll
  for (int kk = 0; kk < 6; ++kk) {
    const int k = 4 * kk + 2 * hf;
    a0[kk].x = ((m0 >> k)       & 1u) ? THIRD : 0.0f;
    a0[kk].y = ((m0 >> (k + 1)) & 1u) ? THIRD : 0.0f;
    a1[kk].x = ((m1 >> k)       & 1u) ? THIRD : 0.0f;
    a1[kk].y = ((m1 >> (k + 1)) & 1u) ? THIRD : 0.0f;
  }

  const long lt0 = (long)widx * TILES_PER_WAVE;              // first 16-wide tile
  // Load base: fold the 2*hf*L row term so all 12 loads are base + immediate.
  const float* __restrict__ p = x + (long)slice * V * L + lt0 * 16 + col + (long)(2 * hf) * L;
  // Store bases (loop-invariant, per-lane):
  //   acc0 VGPR r -> t = r + 8*hf          => po0 + r*L
  //   acc1 VGPR r -> t = 16 + r + 7*hf     => po1 + r*L  (r = 0..6; VGPR 7 dead)
  const long ob = (long)slice * T * L + lt0 * 16 + col;
  float* __restrict__ po0 = out + ob + (long)(8 * hf) * L;
  float* __restrict__ po1 = out + ob + (long)(16 + 7 * hf) * L;

#pragma unroll 1
  for (int i = 0; i < TILES_PER_WAVE; ++i) {
    // B fragments: 4x16 f32; lane n<16 holds rows {4kk,4kk+1}, lanes 16-31 rows {+2,+3}
    // (the +2 rows are already folded into p).
    v2f bv[6];
#pragma unroll
    for (int kk = 0; kk < 6; ++kk) {
      bv[kk].x = p[(long)(4 * kk)     * L];
      bv[kk].y = p[(long)(4 * kk + 1) * L];
    }

    v8f acc0 = {};   // triangles 0..15
    v8f acc1 = {};   // triangles 16..29 (slots 7,15 dead)
#pragma unroll
    for (int kk = 0; kk < 6; ++kk) {
      acc0 = __builtin_amdgcn_wmma_f32_16x16x4_f32(false, a0[kk], false, bv[kk],
                                                   (short)0, acc0, false, false);
      acc1 = __builtin_amdgcn_wmma_f32_16x16x4_f32(false, a1[kk], false, bv[kk],
                                                   (short)0, acc1, false, false);
    }

    // 15 unconditional stores, uniform immediate offsets, no divergence.
#pragma unroll
    for (int r = 0; r < 8; ++r) {
      po0[(long)r * L] = acc0[r];
    }
#pragma unroll
    for (int r = 0; r < 7; ++r) {
      po1[(long)r * L] = acc1[r];
    }

    p   += 16;
    po0 += 16;
    po1 += 16;
  }
}

extern "C" void kernel_launch(void* const* d_in, const int* in_sizes, int n_in,
                              void* d_out, int out_size, void* d_ws, size_t ws_size,
                              hipStream_t stream) {
  (void)in_sizes; (void)n_in; (void)out_size; (void)d_ws; (void)ws_size;
  const float* x   = (const float*)d_in[0];
  float*       out = (float*)d_out;

  // 64 slices * 2048 tiles = 131072 wave-tiles; 8 tiles/wave -> 16384 waves
  // -> 2048 blocks of 256 threads (8 wave32s). Exact coverage: no partial waves,
  // EXEC is all-ones at every WMMA as the ISA requires.
  dim3 grid(2048), block(256);
  tri_mean_wmma<<<grid, block, 0, stream>>>(x, out);
}